// CaptionModel_30305289240920
// MI455X (gfx1250) — compile-verified
//
#include <hip/hip_runtime.h>
#include <hip/hip_bf16.h>

typedef float v2f __attribute__((ext_vector_type(2)));
typedef float v8f __attribute__((ext_vector_type(8)));

#define B_   64
#define N_   49
#define F_   2048
#define H_   512
#define E_   256
#define V_   30522
#define T_   99
#define VPAD 30528          // 1908 * 16
#define XDIM (E_ + F_)      // 2304
#define G4H  (4 * H_)       // 2048

// ---------------------------------------------------------------------------
// fp32 WMMA 16x16x4 fragment helpers (wave32).
// A (16x4, row-major src): lane<16 -> K=k0,k0+1 ; lane>=16 -> K=k0+2,k0+3
// B (4x16, row-major src): lane%16 = column; same K split as A.
// C/D: vgpr r, lane L -> row = r + (L>=16)*8, col = L%16
// ---------------------------------------------------------------------------
__device__ __forceinline__ v2f load_a_frag(const float* __restrict__ A, int lda,
                                           int row0, int k0, int lane) {
  int row = row0 + (lane & 15);
  int kk  = k0 + ((lane >> 4) << 1);
  const float* p = A + (size_t)row * lda + kk;
  v2f a; a.x = p[0]; a.y = p[1];
  return a;
}
__device__ __forceinline__ v2f load_b_frag(const float* __restrict__ Bm, int ldb,
                                           int k0, int col0, int lane) {
  int col = col0 + (lane & 15);
  int kk  = k0 + ((lane >> 4) << 1);
  const float* p = Bm + (size_t)kk * ldb + col;
  v2f b; b.x = p[0]; b.y = p[ldb];
  return b;
}

__device__ __forceinline__ v8f gemm_acc(const float* __restrict__ A, int lda,
                                        const float* __restrict__ Bm, int ldb,
                                        int row0, int col0, int K, v8f acc, int lane) {
  for (int k = 0; k < K; k += 4) {
    v2f a = load_a_frag(A, lda, row0, k, lane);
    v2f b = load_b_frag(Bm, ldb, k, col0, lane);
    acc = __builtin_amdgcn_wmma_f32_16x16x4_f32(false, a, false, b,
                                                (short)0, acc, false, false);
  }
  return acc;
}

__device__ __forceinline__ float sigmoidf_(float v) { return 1.f / (1.f + __expf(-v)); }

// ---------------------------------------------------------------------------
// init: h0 = c0 = 0 ; x[:, :E] = embed[sos]
// ---------------------------------------------------------------------------
__global__ void init_kernel(const int* __restrict__ sos,
                            const float* __restrict__ embed,
                            float* __restrict__ h0, float* __restrict__ c0,
                            float* __restrict__ x) {
  int b = blockIdx.x, tid = threadIdx.x;           // blockDim = 512
  h0[b * H_ + tid] = 0.f;
  c0[b * H_ + tid] = 0.f;
  if (tid < E_) x[(size_t)b * XDIM + tid] = embed[(size_t)sos[0] * E_ + tid];
}

// ---------------------------------------------------------------------------
// feat_proj[BN,H] = features[BN,F] @ Wf[F,H]     (one-time)
// ---------------------------------------------------------------------------
__global__ void featproj_kernel(const float* __restrict__ features,
                                const float* __restrict__ Wf,
                                float* __restrict__ fp) {
  int lane = threadIdx.x;
  int row0 = blockIdx.x * 16, col0 = blockIdx.y * 16;
  v8f acc = {};
  acc = gemm_acc(features, F_, Wf, H_, row0, col0, F_, acc, lane);
  int col = col0 + (lane & 15);
  int rbase = row0 + ((lane >> 4) << 3);
  for (int r = 0; r < 8; ++r) fp[(size_t)(rbase + r) * H_ + col] = acc[r];
}

// ---------------------------------------------------------------------------
// hW[B,H] = h[B,H] @ attn_Wh[H,H] + attn_b
// ---------------------------------------------------------------------------
__global__ void hw_kernel(const float* __restrict__ h, const float* __restrict__ Wh,
                          const float* __restrict__ bias, float* __restrict__ hW) {
  int lane = threadIdx.x;
  int row0 = blockIdx.x * 16, col0 = blockIdx.y * 16;
  int col = col0 + (lane & 15);
  float bv = bias[col];
  v8f acc; for (int r = 0; r < 8; ++r) acc[r] = bv;
  acc = gemm_acc(h, H_, Wh, H_, row0, col0, H_, acc, lane);
  int rbase = row0 + ((lane >> 4) << 3);
  for (int r = 0; r < 8; ++r) hW[(size_t)(rbase + r) * H_ + col] = acc[r];
}

// ---------------------------------------------------------------------------
// per-batch attention: scores -> softmax -> context into x[:, E:]
// ---------------------------------------------------------------------------
__global__ void attn_ctx_kernel(const float* __restrict__ fp,       // [B*N,H]
                                const float* __restrict__ hW,       // [B,H]
                                const float* __restrict__ attn_v,   // [H]
                                const float* __restrict__ features, // [B,N,F]
                                float* __restrict__ x) {            // [B,XDIM]
  __shared__ float s_hw[H_];
  __shared__ float s_v[H_];
  __shared__ float s_w[N_];
  int b = blockIdx.x, tid = threadIdx.x;           // blockDim = 256
  for (int i = tid; i < H_; i += 256) { s_hw[i] = hW[(size_t)b * H_ + i]; s_v[i] = attn_v[i]; }
  __syncthreads();
  int wave = tid >> 5, lane = tid & 31;
  for (int n = wave; n < N_; n += 8) {
    const float* fpn = fp + ((size_t)b * N_ + n) * H_;
    float sum = 0.f;
    for (int hh = lane; hh < H_; hh += 32)
      sum += tanhf(fpn[hh] + s_hw[hh]) * s_v[hh];
    for (int off = 16; off > 0; off >>= 1) sum += __shfl_xor(sum, off, 32);
    if (lane == 0) s_w[n] = sum;
  }
  __syncthreads();
  if (tid == 0) {
    float mx = s_w[0];
    for (int n = 1; n < N_; ++n) mx = fmaxf(mx, s_w[n]);
    float ssum = 0.f;
    for (int n = 0; n < N_; ++n) { float e = __expf(s_w[n] - mx); s_w[n] = e; ssum += e; }
    float inv = 1.f / ssum;
    for (int n = 0; n < N_; ++n) s_w[n] *= inv;
  }
  __syncthreads();
  const float* fb = features + (size_t)b * N_ * F_;
  for (int f = tid; f < F_; f += 256) {
    float acc = 0.f;
    for (int n = 0; n < N_; ++n) acc += s_w[n] * fb[(size_t)n * F_ + f];
    x[(size_t)b * XDIM + E_ + f] = acc;
  }
}

// ---------------------------------------------------------------------------
// fused LSTM: gates = x@W_ih + b_ih + h@W_hh + b_hh, then cell update.
// One wave per 16x16 hidden tile, 4 accumulators (i,f,g,o) share A fragments.
// ---------------------------------------------------------------------------
__global__ void lstm_kernel(const float* __restrict__ x,     // [B,XDIM]
                            const float* __restrict__ W_ih,  // [XDIM,4H]
                            const float* __restrict__ b_ih,  // [4H]
                            const float* __restrict__ W_hh,  // [H,4H]
                            const float* __restrict__ b_hh,  // [4H]
                            const float* __restrict__ h_in,  // [B,H]
                            const float* __restrict__ c_in,  // [B,H]
                            float* __restrict__ h_out,
                            float* __restrict__ c_out) {
  int lane = threadIdx.x;
  int row0 = blockIdx.x * 16, j0 = blockIdx.y * 16;
  int col = j0 + (lane & 15);
  v8f acc[4];
  for (int g = 0; g < 4; ++g) {
    float bv = b_ih[g * H_ + col] + b_hh[g * H_ + col];
    for (int r = 0; r < 8; ++r) acc[g][r] = bv;
  }
  for (int k = 0; k < XDIM; k += 4) {
    v2f a = load_a_frag(x, XDIM, row0, k, lane);
    for (int g = 0; g < 4; ++g) {
      v2f bf = load_b_frag(W_ih, G4H, k, g * H_ + j0, lane);
      acc[g] = __builtin_amdgcn_wmma_f32_16x16x4_f32(false, a, false, bf,
                                                     (short)0, acc[g], false, false);
    }
  }
  for (int k = 0; k < H_; k += 4) {
    v2f a = load_a_frag(h_in, H_, row0, k, lane);
    for (int g = 0; g < 4; ++g) {
      v2f bf = load_b_frag(W_hh, G4H, k, g * H_ + j0, lane);
      acc[g] = __builtin_amdgcn_wmma_f32_16x16x4_f32(false, a, false, bf,
                                                     (short)0, acc[g], false, false);
    }
  }
  int rbase = row0 + ((lane >> 4) << 3);
  for (int r = 0; r < 8; ++r) {
    int m = rbase + r;
    float iv = sigmoidf_(acc[0][r]);
    float fv = sigmoidf_(acc[1][r]);
    float gv = tanhf(acc[2][r]);
    float ov = sigmoidf_(acc[3][r]);
    float cn = fv * c_in[(size_t)m * H_ + col] + iv * gv;
    float hn = ov * tanhf(cn);
    c_out[(size_t)m * H_ + col] = cn;
    h_out[(size_t)m * H_ + col] = hn;
  }
}

// ---------------------------------------------------------------------------
// logits = h @ W_proj + b_proj ; write out[B,V,T] slice t and logits scratch.
// ---------------------------------------------------------------------------
__global__ void proj_kernel(const float* __restrict__ h,   // [B,H]
                            const float* __restrict__ Wp,  // [H,V]
                            const float* __restrict__ bp,  // [V]
                            float* __restrict__ logits,    // [B,VPAD]
                            float* __restrict__ out,       // [B,V,T]
                            int t) {
  int lane = threadIdx.x;
  int row0 = blockIdx.x * 16, col0 = blockIdx.y * 16;
  int col = col0 + (lane & 15);
  bool cok = col < V_;
  float bv = cok ? bp[col] : 0.f;
  v8f acc; for (int r = 0; r < 8; ++r) acc[r] = bv;
  for (int k = 0; k < H_; k += 4) {
    v2f a = load_a_frag(h, H_, row0, k, lane);
    int kk = k + ((lane >> 4) << 1);
    v2f bf;
    bf.x = cok ? Wp[(size_t)kk * V_ + col] : 0.f;
    bf.y = cok ? Wp[(size_t)(kk + 1) * V_ + col] : 0.f;
    acc = __builtin_amdgcn_wmma_f32_16x16x4_f32(false, a, false, bf,
                                                (short)0, acc, false, false);
  }
  int rbase = row0 + ((lane >> 4) << 3);
  if (cok) {
    for (int r = 0; r < 8; ++r) {
      int m = rbase + r;
      float v = acc[r];
      logits[(size_t)m * VPAD + col] = v;
      out[((size_t)m * V_ + col) * T_ + t] = v;
    }
  }
}

// ---------------------------------------------------------------------------
// greedy argmax over V per batch row, then gather next embedding into x[:, :E]
// ---------------------------------------------------------------------------
__global__ void argmax_embed_kernel(const float* __restrict__ logits, // [B,VPAD]
                                    const float* __restrict__ embed,  // [V,E]
                                    float* __restrict__ x) {          // [B,XDIM]
  __shared__ float s_val[256];
  __shared__ int   s_idx[256];
  int b = blockIdx.x, tid = threadIdx.x;           // blockDim = 256 == E_
  const float* lg = logits + (size_t)b * VPAD;
  float best = -3.402823466e38f; int bi = 0;
  for (int v = tid; v < V_; v += 256) {
    float lv = lg[v];
    if (lv > best) { best = lv; bi = v; }
  }
  s_val[tid] = best; s_idx[tid] = bi;
  __syncthreads();
  for (int s = 128; s > 0; s >>= 1) {
    if (tid < s) {
      float ov = s_val[tid + s]; int oi = s_idx[tid + s];
      if (ov > s_val[tid] || (ov == s_val[tid] && oi < s_idx[tid])) {
        s_val[tid] = ov; s_idx[tid] = oi;
      }
    }
    __syncthreads();
  }
  int idx = s_idx[0];
  x[(size_t)b * XDIM + tid] = embed[(size_t)idx * E_ + tid];
}

// ---------------------------------------------------------------------------
extern "C" void kernel_launch(void* const* d_in, const int* in_sizes, int n_in,
                              void* d_out, int out_size, void* d_ws, size_t ws_size,
                              hipStream_t stream) {
  (void)in_sizes; (void)n_in; (void)out_size; (void)ws_size;
  const float* features = (const float*)d_in[0];
  const float* embed    = (const float*)d_in[1];
  const float* attn_Wf  = (const float*)d_in[2];
  const float* attn_Wh  = (const float*)d_in[3];
  const float* attn_b   = (const float*)d_in[4];
  const float* attn_v   = (const float*)d_in[5];
  const float* W_ih     = (const float*)d_in[6];
  const float* b_ih     = (const float*)d_in[7];
  const float* W_hh     = (const float*)d_in[8];
  const float* b_hh     = (const float*)d_in[9];
  const float* W_proj   = (const float*)d_in[10];
  const float* b_proj   = (const float*)d_in[11];
  const int*   sos      = (const int*)d_in[12];
  float* out = (float*)d_out;

  float* ws = (float*)d_ws;
  size_t off = 0;
  float* fp     = ws + off; off += (size_t)B_ * N_ * H_;   // 1,605,632
  float* hW     = ws + off; off += (size_t)B_ * H_;
  float* hbuf0  = ws + off; off += (size_t)B_ * H_;
  float* hbuf1  = ws + off; off += (size_t)B_ * H_;
  float* cbuf0  = ws + off; off += (size_t)B_ * H_;
  float* cbuf1  = ws + off; off += (size_t)B_ * H_;
  float* x      = ws + off; off += (size_t)B_ * XDIM;
  float* logits = ws + off; off += (size_t)B_ * VPAD;      // total ~15.5 MB

  init_kernel<<<dim3(B_), dim3(H_), 0, stream>>>(sos, embed, hbuf0, cbuf0, x);
  featproj_kernel<<<dim3((B_ * N_) / 16, H_ / 16), dim3(32), 0, stream>>>(
      features, attn_Wf, fp);

  for (int t = 0; t < T_; ++t) {
    float* h_cur = (t & 1) ? hbuf1 : hbuf0;
    float* c_cur = (t & 1) ? cbuf1 : cbuf0;
    float* h_nxt = (t & 1) ? hbuf0 : hbuf1;
    float* c_nxt = (t & 1) ? cbuf0 : cbuf1;

    hw_kernel<<<dim3(B_ / 16, H_ / 16), dim3(32), 0, stream>>>(
        h_cur, attn_Wh, attn_b, hW);
    attn_ctx_kernel<<<dim3(B_), dim3(256), 0, stream>>>(
        fp, hW, attn_v, features, x);
    lstm_kernel<<<dim3(B_ / 16, H_ / 16), dim3(32), 0, stream>>>(
        x, W_ih, b_ih, W_hh, b_hh, h_cur, c_cur, h_nxt, c_nxt);
    proj_kernel<<<dim3(B_ / 16, VPAD / 16), dim3(32), 0, stream>>>(
        h_nxt, W_proj, b_proj, logits, out, t);
    if (t + 1 < T_)
      argmax_embed_kernel<<<dim3(B_), dim3(256), 0, stream>>>(logits, embed, x);
  }
}